// TreeNetworkExtra_22789096473145
// MI455X (gfx1250) — compile-verified
//
#include <hip/hip_runtime.h>
#include <math.h>

// ---------------- constants ----------------
#define CBQ   256      // CB
#define CEQ   64       // CE
#define GQ    16       // G
#define BQ    2        // batch
#define INFW  1e30f
#define NMAXQ 7680     // 48*160 (largest level grid)
#define NUPQ  30720    // 96*320 (after final upsample)
#define MBLK  4        // M-tiles (16 rows each) per wave: 64x16 output slab

typedef float v2f __attribute__((ext_vector_type(2)));
typedef float v8f __attribute__((ext_vector_type(8)));

// fp32 WMMA: D(16x16) += A(16x4) * B(4x16), full fp32 precision (matches reference dtype)
__device__ __forceinline__ v8f wmma4(v2f a, v2f b, v8f c) {
  return __builtin_amdgcn_wmma_f32_16x16x4_f32(false, a, false, b, (short)0, c,
                                               false, false);
}

// ---------------- GEMM (1x1 conv): D[co][n] = sum_ci W[co][ci] * X[ci][n] ----------------
// mode 0: plain   mode 1: v*g[m]+b[m], relu   mode 2: relu(v + R[m][n])
// grid: (ceil(N/16), CO/64, B)  block: 32. One wave computes a 64x16 slab (4 WMMA
// accumulators) so the B operand is loaded ONCE per k-step and reused 4x.
// Out-of-range columns are CLAMPED (not predicated): WMMA column n only feeds output
// column n, and clamped columns are never stored -> unconditional non-divergent loads.
__global__ __launch_bounds__(32)
void gemm1x1_wmma(const float* __restrict__ Wt, const float* __restrict__ X,
                  float* __restrict__ D, int CO, int CI, int N, int mode,
                  const float* __restrict__ gg, const float* __restrict__ bb,
                  const float* __restrict__ R) {
  int lane = threadIdx.x;
  int tn = blockIdx.x * 16, tm = blockIdx.y * (16 * MBLK);
  size_t xoff = (size_t)blockIdx.z * CI * N;
  size_t doff = (size_t)blockIdx.z * CO * N;
  int col  = tn + (lane & 15);
  int colC = col < N ? col : N - 1;      // clamp, keep loads unconditional
  int koff = (lane >> 4) * 2;            // lanes 0-15: K offs 0,1 ; lanes 16-31: 2,3
  const float* Wr[MBLK];
  #pragma unroll
  for (int t = 0; t < MBLK; ++t)
    Wr[t] = Wt + (size_t)(tm + t * 16 + (lane & 15)) * CI;
  const float* Xc = X + xoff + colC;
  v8f acc[MBLK] = {};
  for (int k = 0; k < CI; k += 4) {
    v2f b;
    b.x = Xc[(size_t)(k + koff) * N];
    b.y = Xc[(size_t)(k + koff + 1) * N];
    #pragma unroll
    for (int t = 0; t < MBLK; ++t) {
      v2f a;
      a.x = Wr[t][k + koff];
      a.y = Wr[t][k + koff + 1];
      acc[t] = wmma4(a, b, acc[t]);
    }
  }
  if (col >= N) return;
  #pragma unroll
  for (int t = 0; t < MBLK; ++t) {
    int mBase = tm + t * 16 + ((lane >> 4) * 8);  // VGPR r -> row r / r+8
    #pragma unroll
    for (int r = 0; r < 8; ++r) {
      int m = mBase + r;
      float v = acc[t][r];
      if (mode == 1)       v = fmaxf(fmaf(v, gg[m], bb[m]), 0.f);
      else if (mode == 2)  v = fmaxf(v + R[doff + (size_t)m * N + col], 0.f);
      D[doff + (size_t)m * N + col] = v;
    }
  }
}

// ---------------- GEMM (3x3 conv, implicit im2col, pad=1) ----------------
// K = CI*9; k -> (ci, tap). Zero padding realized as an ADDRESS SELECT against a
// zeroed scratch word (zpage): v_cndmask on the address + plain load, instead of
// divergent exec-masked loads. 64x16 slab per wave: the expensive gathered B operand
// (and its address math) is computed once per k-step, reused by 4 WMMAs.
__global__ __launch_bounds__(32)
void gemm3x3_wmma(const float* __restrict__ Wt, const float* __restrict__ X,
                  float* __restrict__ D, int CO, int CI, int H, int Wd, int mode,
                  const float* __restrict__ gg, const float* __restrict__ bb,
                  const float* __restrict__ R, const float* __restrict__ zpage) {
  int lane = threadIdx.x;
  int N = H * Wd;
  int K = CI * 9;
  int tn = blockIdx.x * 16, tm = blockIdx.y * (16 * MBLK);
  size_t xoff = (size_t)blockIdx.z * CI * N;
  size_t doff = (size_t)blockIdx.z * CO * N;
  int col  = tn + (lane & 15);
  int colC = col < N ? col : N - 1;
  int h0 = colC / Wd;
  int w0 = colC - h0 * Wd;
  int koff = (lane >> 4) * 2;
  const float* Wr[MBLK];
  #pragma unroll
  for (int t = 0; t < MBLK; ++t)
    Wr[t] = Wt + (size_t)(tm + t * 16 + (lane & 15)) * K;
  const float* Xb = X + xoff;
  v8f acc[MBLK] = {};
  for (int k = 0; k < K; k += 4) {
    v2f b;
    #pragma unroll
    for (int j = 0; j < 2; ++j) {
      int kk  = k + koff + j;
      int ci  = kk / 9;
      int tap = kk - ci * 9;
      int h = h0 + tap / 3 - 1;
      int w = w0 + (tap % 3) - 1;
      bool ok = ((unsigned)h < (unsigned)H) & ((unsigned)w < (unsigned)Wd);
      const float* p = ok ? (Xb + ((size_t)ci * H + h) * Wd + w) : zpage;
      float bv = *p;                      // always issued; zero page supplies padding
      if (j == 0) b.x = bv; else b.y = bv;
    }
    #pragma unroll
    for (int t = 0; t < MBLK; ++t) {
      v2f a;
      a.x = Wr[t][k + koff];
      a.y = Wr[t][k + koff + 1];
      acc[t] = wmma4(a, b, acc[t]);
    }
  }
  if (col >= N) return;
  #pragma unroll
  for (int t = 0; t < MBLK; ++t) {
    int mBase = tm + t * 16 + ((lane >> 4) * 8);
    #pragma unroll
    for (int r = 0; r < 8; ++r) {
      int m = mBase + r;
      float v = acc[t][r];
      if (mode == 1)       v = fmaxf(fmaf(v, gg[m], bb[m]), 0.f);
      else if (mode == 2)  v = fmaxf(v + R[doff + (size_t)m * N + col], 0.f);
      D[doff + (size_t)m * N + col] = v;
    }
  }
}

// ---------------- global average pool over H*W ----------------
__global__ void gap_kernel(const float* __restrict__ x, float* __restrict__ out,
                           int C, int HW) {
  int id = blockIdx.x * blockDim.x + threadIdx.x;   // b*C + c
  if (id >= BQ * C) return;
  const float* p = x + (size_t)id * HW;
  float s = 0.f;
  for (int i = 0; i < HW; ++i) s += p[i];
  out[id] = s / (float)HW;
}

// gc: relu(g*(W(256x2048) . gap) + b) — one block per (b,co)
__global__ __launch_bounds__(256)
void gc_kernel(const float* __restrict__ gap, const float* __restrict__ Wt,
               const float* __restrict__ gg, const float* __restrict__ bb,
               float* __restrict__ out, int CI) {
  __shared__ float red[256];
  int id = blockIdx.x;
  int b = id / CBQ, co = id - b * CBQ;
  const float* gp = gap + (size_t)b * CI;
  const float* wr = Wt + (size_t)co * CI;
  float s = 0.f;
  for (int c = threadIdx.x; c < CI; c += 256) s += wr[c] * gp[c];
  red[threadIdx.x] = s; __syncthreads();
  for (int st = 128; st > 0; st >>= 1) {
    if (threadIdx.x < st) red[threadIdx.x] += red[threadIdx.x + st];
    __syncthreads();
  }
  if (threadIdx.x == 0) out[id] = fmaxf(fmaf(red[0], gg[co], bb[co]), 0.f);
}

__global__ void broadcast_kernel(const float* __restrict__ v, float* __restrict__ out, int N) {
  int id = blockIdx.x * blockDim.x + threadIdx.x;
  if (id >= BQ * CBQ * N) return;
  out[id] = v[id / N];
}

// ---------------- MST edge weights: w[n][d] = sum_c (f[c][n]-f[c][nbr])^2 ----------------
__global__ void edgew_kernel(const float* __restrict__ fm, float* __restrict__ wE,
                             int C, int H, int Wd) {
  int N = H * Wd;
  int id = blockIdx.x * blockDim.x + threadIdx.x;   // n*4 + d
  if (id >= N * 4) return;
  int n = id >> 2, d = id & 3;
  int h = n / Wd, w = n - h * Wd;
  int v = -1;
  if (d == 0)      { if (w > 0)      v = n - 1;  }   // left
  else if (d == 1) { if (w < Wd - 1) v = n + 1;  }   // right
  else if (d == 2) { if (h > 0)      v = n - Wd; }   // up
  else             { if (h < H - 1)  v = n + Wd; }   // down
  float s;
  if (v < 0) s = INFW;
  else {
    s = 0.f;
    const float* p = fm + (size_t)blockIdx.z * C * N;
    for (int c = 0; c < C; ++c) {
      float dd = p[(size_t)c * N + n] - p[(size_t)c * N + v];
      s += dd * dd;
    }
  }
  wE[(size_t)blockIdx.z * N * 4 + id] = s;
}

// ---------------- Prim MST, one block per image, all state in LDS (CDNA5: 320KB/WGP) ----
__global__ __launch_bounds__(256)
void prim_kernel(const float* __restrict__ wE, int H, int Wd,
                 int* __restrict__ orderG, int* __restrict__ parentG) {
  __shared__ float key[NMAXQ];
  __shared__ int   par[NMAXQ];
  __shared__ unsigned char intree[NMAXQ];
  __shared__ float rv[256];
  __shared__ int   ri[256];
  int N = H * Wd;
  int b = blockIdx.x;
  int t = threadIdx.x;
  const float* wb = wE + (size_t)b * N * 4;
  for (int n = t; n < N; n += 256) { key[n] = INFW; par[n] = 0; intree[n] = 0; }
  __syncthreads();
  if (t == 0) key[0] = 0.f;
  __syncthreads();
  for (int step = 0; step < N; ++step) {
    // argmin over !intree (ties -> lowest index, matching jnp.argmin)
    float bv = INFW; int bi = N;
    for (int n = t; n < N; n += 256) {
      float v = intree[n] ? INFW : key[n];
      if (v < bv) { bv = v; bi = n; }
    }
    rv[t] = bv; ri[t] = bi;
    __syncthreads();
    for (int st = 128; st > 0; st >>= 1) {
      if (t < st) {
        if (rv[t + st] < rv[t] || (rv[t + st] == rv[t] && ri[t + st] < ri[t])) {
          rv[t] = rv[t + st]; ri[t] = ri[t + st];
        }
      }
      __syncthreads();
    }
    int u = ri[0];
    if (t == 0) { intree[u] = 1; orderG[(size_t)b * N + step] = u; }
    __syncthreads();
    if (t < 4) {                      // relax the 4 grid neighbors (distinct v: no race)
      int h = u / Wd, w = u - h * Wd;
      int v = -1;
      if (t == 0)      { if (w > 0)      v = u - 1;  }
      else if (t == 1) { if (w < Wd - 1) v = u + 1;  }
      else if (t == 2) { if (h > 0)      v = u - Wd; }
      else             { if (h < H - 1)  v = u + Wd; }
      if (v >= 0 && !intree[v]) {
        float ww = wb[(size_t)u * 4 + t];
        if (ww < key[v]) { key[v] = ww; par[v] = u; }
      }
    }
    __syncthreads();
  }
  for (int n = t; n < N; n += 256) parentG[(size_t)b * N + n] = par[n];
}

// ---------------- S[g][n] = exp(-||e_g[n] - e_g[parent[n]]||^2), S[:,0]=0 ----------------
__global__ void s_kernel(const float* __restrict__ embed, const int* __restrict__ parent,
                         float* __restrict__ S, int N) {
  int id = blockIdx.x * blockDim.x + threadIdx.x;   // g*N + n
  if (id >= GQ * N) return;
  int g = id / N, n = id - g * N;
  int b = blockIdx.z;
  float s;
  if (n == 0) s = 0.f;
  else {
    int p = parent[(size_t)b * N + n];
    const float* e = embed + (size_t)b * CEQ * N;
    float acc = 0.f;
    for (int c = 0; c < CEQ / GQ; ++c) {
      const float* ec = e + (size_t)(g * (CEQ / GQ) + c) * N;
      float d = ec[n] - ec[p];
      acc += d * d;
    }
    s = expf(-acc);
  }
  S[((size_t)b * GQ + g) * N + n] = s;
}

// ---------------- tree filter: up + down scans. One wave per (image, group). ----------------
// Each of the 17 channels (16 data + denominator-of-ones) is an independent serial
// recurrence -> one lane per channel, zero intra-scan synchronization.
__global__ __launch_bounds__(32)
void treefilter_kernel(const float* __restrict__ last, const float* __restrict__ S,
                       const int* __restrict__ orderG, const int* __restrict__ parentG,
                       float* __restrict__ aggr, float* __restrict__ filt, int N) {
  int b = blockIdx.z, g = blockIdx.x, c = threadIdx.x;
  const float* Sg  = S + ((size_t)b * GQ + g) * N;
  const int*   ord = orderG  + (size_t)b * N;
  const int*   par = parentG + (size_t)b * N;
  float* A = aggr + (((size_t)b * GQ + g) * 17 + (c < 17 ? c : 0)) * N;
  if (c < 17) {
    if (c < 16) {
      const float* src = last + ((size_t)b * CBQ + g * 16 + c) * N;
      for (int n = 0; n < N; ++n) A[n] = src[n];
    } else {
      for (int n = 0; n < N; ++n) A[n] = 1.0f;
    }
    // up: leaves -> root
    for (int i = N - 1; i >= 0; --i) { int u = ord[i]; A[par[u]] += Sg[u] * A[u]; }
    // down: root -> leaves (in place; parent finalized before child in Prim order)
    for (int i = 0; i < N; ++i) {
      int u = ord[i]; int p = par[u]; float s = Sg[u];
      float au = A[u];
      A[u] = au + s * (A[p] - s * au);
    }
  }
  __syncthreads();
  if (c < 16) {
    const float* Aden = aggr + (((size_t)b * GQ + g) * 17 + 16) * N;
    float* F = filt + ((size_t)b * CBQ + g * 16 + c) * N;
    for (int n = 0; n < N; ++n) F[n] = A[n] / Aden[n];
  }
}

__global__ void add_kernel(float* __restrict__ a, const float* __restrict__ b, int total) {
  int id = blockIdx.x * blockDim.x + threadIdx.x;
  if (id < total) a[id] += b[id];
}

// ---------------- align-corners bilinear upsample ----------------
__global__ void upsample_kernel(const float* __restrict__ x, float* __restrict__ y,
                                int H, int Wd, int Ho, int Wo) {
  int id = blockIdx.x * blockDim.x + threadIdx.x;
  if (id >= BQ * CBQ * Ho * Wo) return;
  int wo = id % Wo;
  int tmp = id / Wo;
  int ho = tmp % Ho;
  int bc = tmp / Ho;
  float sy = (float)(H - 1) / (float)(Ho - 1);
  float sx = (float)(Wd - 1) / (float)(Wo - 1);
  float ys = ho * sy, xs = wo * sx;
  int y0 = (int)floorf(ys); if (y0 > H - 2) y0 = H - 2; if (y0 < 0) y0 = 0;
  int x0 = (int)floorf(xs); if (x0 > Wd - 2) x0 = Wd - 2; if (x0 < 0) x0 = 0;
  int y1 = (y0 + 1 < H)  ? y0 + 1 : H - 1;
  int x1 = (x0 + 1 < Wd) ? x0 + 1 : Wd - 1;
  float wy = ys - y0, wx = xs - x0;
  const float* p = x + (size_t)bc * H * Wd;
  float v00 = p[y0 * Wd + x0], v01 = p[y0 * Wd + x1];
  float v10 = p[y1 * Wd + x0], v11 = p[y1 * Wd + x1];
  float r0 = v00 * (1.f - wy) + v10 * wy;
  float r1 = v01 * (1.f - wy) + v11 * wy;
  y[id] = r0 * (1.f - wx) + r1 * wx;
}

// ---------------- final head: sigmoid(w.t + b) scaled to [0.01, 10] ----------------
__global__ void head_kernel(const float* __restrict__ t, const float* __restrict__ wv,
                            const float* __restrict__ bv, float* __restrict__ out, int N) {
  int id = blockIdx.x * blockDim.x + threadIdx.x;
  if (id >= BQ * N) return;
  int b = id / N, n = id - b * N;
  const float* p = t + (size_t)b * CBQ * N + n;
  float s = bv[0];
  for (int c = 0; c < CBQ; ++c) s += wv[c] * p[(size_t)c * N];
  float sig = 1.f / (1.f + expf(-s));
  out[id] = 0.01f + 9.99f * sig;
}

// ==================== host orchestration ====================
extern "C" void kernel_launch(void* const* d_in, const int* in_sizes, int n_in,
                              void* d_out, int out_size, void* d_ws, size_t ws_size,
                              hipStream_t stream) {
  (void)in_sizes; (void)n_in; (void)out_size; (void)ws_size;
  const float* b_in[4] = {(const float*)d_in[0], (const float*)d_in[1],
                          (const float*)d_in[2], (const float*)d_in[3]};
  // jax pytree flatten order (dict keys sorted): gc(b,g,w), head0-2, level0-3
  const float* gc_b = (const float*)d_in[4];
  const float* gc_g = (const float*)d_in[5];
  const float* gc_w = (const float*)d_in[6];
  // head i at 7+7*i: [b, rr.cbr.b, rr.cbr.g, rr.cbr.w, rr.w1, rr.w2, w]; only head2 feeds output
  const float* h2_b     = (const float*)d_in[21];
  const float* h2_cbr_b = (const float*)d_in[22];
  const float* h2_cbr_g = (const float*)d_in[23];
  const float* h2_cbr_w = (const float*)d_in[24];
  const float* h2_w1    = (const float*)d_in[25];
  const float* h2_w2    = (const float*)d_in[26];
  const float* h2_w     = (const float*)d_in[27];
  // level i at 28+11*i: [embed_w, lat.cbr.b, lat.cbr.g, lat.cbr.w, lat.w1, lat.w2,
  //                      ref.cbr.b, ref.cbr.g, ref.cbr.w, ref.w1, ref.w2]
  const float *embW[4], *Lcb[4], *Lcg[4], *Lcw[4], *Lw1[4], *Lw2[4],
              *Rcb[4], *Rcg[4], *Rcw[4], *Rw1[4], *Rw2[4];
  for (int i = 0; i < 4; ++i) {
    int base = 28 + 11 * i;
    embW[i] = (const float*)d_in[base + 0];
    Lcb[i]  = (const float*)d_in[base + 1];
    Lcg[i]  = (const float*)d_in[base + 2];
    Lcw[i]  = (const float*)d_in[base + 3];
    Lw1[i]  = (const float*)d_in[base + 4];
    Lw2[i]  = (const float*)d_in[base + 5];
    Rcb[i]  = (const float*)d_in[base + 6];
    Rcg[i]  = (const float*)d_in[base + 7];
    Rcw[i]  = (const float*)d_in[base + 8];
    Rw1[i]  = (const float*)d_in[base + 9];
    Rw2[i]  = (const float*)d_in[base + 10];
  }

  // workspace bump allocator
  char* ws = (char*)d_ws;
  size_t off = 0;
  auto carve = [&](size_t bytes) -> void* {
    void* p = ws + off;
    off += (bytes + 255) & ~(size_t)255;
    return p;
  };
  float* zpage = (float*)carve(256);              // zero page for padded loads
  float* last = (float*)carve((size_t)BQ * CBQ * NUPQ * 4);
  float* xbuf = (float*)carve((size_t)BQ * CBQ * NUPQ * 4);
  float* t1   = (float*)carve((size_t)BQ * CBQ * NUPQ * 4);
  size_t reuse_mark = off;                        // level-only buffers; reused by head
  float* lat  = (float*)carve((size_t)BQ * CBQ * NMAXQ * 4);
  float* ref  = (float*)carve((size_t)BQ * CBQ * NMAXQ * 4);
  float* filt = (float*)carve((size_t)BQ * CBQ * NMAXQ * 4);
  float* aggr = (float*)carve((size_t)BQ * GQ * 17 * NMAXQ * 4);
  float* hbuf = (float*)(ws + reuse_mark);        // aliases lat..aggr (63.9MB >= 62.9MB needed)
  float* embedB = (float*)carve((size_t)BQ * CEQ * NMAXQ * 4);
  float* Sbuf   = (float*)carve((size_t)BQ * GQ * NMAXQ * 4);
  float* wE     = (float*)carve((size_t)BQ * NMAXQ * 4 * 4);
  int*   orderB  = (int*)carve((size_t)BQ * NMAXQ * 4);
  int*   parentB = (int*)carve((size_t)BQ * NMAXQ * 4);
  float* gapB = (float*)carve((size_t)BQ * 2048 * 4);
  float* gap2 = (float*)carve((size_t)BQ * CBQ * 4);

  hipMemsetAsync(zpage, 0, 256, stream);          // captured as a graph node; deterministic

  const int Hs[4] = {6, 12, 24, 48};
  const int Wds[4] = {20, 40, 80, 160};
  const int CIN[4] = {2048, 1024, 512, 256};

  // ---- gap -> gc -> broadcast as initial last_fm (6x20) ----
  {
    int total = BQ * 2048;
    gap_kernel<<<(total + 255) / 256, 256, 0, stream>>>(b_in[0], gapB, 2048, 120);
  }
  gc_kernel<<<BQ * CBQ, 256, 0, stream>>>(gapB, gc_w, gc_g, gc_b, gap2, 2048);
  {
    int N0 = 120, total = BQ * CBQ * N0;
    broadcast_kernel<<<(total + 255) / 256, 256, 0, stream>>>(gap2, last, N0);
  }

  // ---- 4 pyramid levels ----
  for (int idx = 0; idx < 4; ++idx) {
    int H = Hs[idx], Wd = Wds[idx], CI = CIN[idx], N = H * Wd;
    const float* fm = b_in[idx];
    dim3 blk(32);
    dim3 g1((N + 15) / 16, CBQ / (16 * MBLK), BQ);
    // latent = refine_residual(fm): x = W1.fm ; t = bnrelu(conv3(x)) ; relu(conv3(t)+x)
    gemm1x1_wmma<<<g1, blk, 0, stream>>>(Lw1[idx], fm, xbuf, CBQ, CI, N, 0,
                                         nullptr, nullptr, nullptr);
    gemm3x3_wmma<<<g1, blk, 0, stream>>>(Lcw[idx], xbuf, t1, CBQ, CBQ, H, Wd, 1,
                                         Lcg[idx], Lcb[idx], nullptr, zpage);
    gemm3x3_wmma<<<g1, blk, 0, stream>>>(Lw2[idx], t1, lat, CBQ, CBQ, H, Wd, 2,
                                         nullptr, nullptr, xbuf, zpage);
    // MST over raw features
    {
      int total = N * 4;
      dim3 gb((total + 255) / 256, 1, BQ);
      edgew_kernel<<<gb, 256, 0, stream>>>(fm, wE, CI, H, Wd);
    }
    prim_kernel<<<BQ, 256, 0, stream>>>(wE, H, Wd, orderB, parentB);
    // embed = 1x1 conv of last_fm (64 out channels -> one 64-row slab)
    {
      dim3 ge((N + 15) / 16, CEQ / (16 * MBLK), BQ);
      gemm1x1_wmma<<<ge, blk, 0, stream>>>(embW[idx], last, embedB, CEQ, CBQ, N, 0,
                                           nullptr, nullptr, nullptr);
    }
    {
      int total = GQ * N;
      dim3 gs((total + 255) / 256, 1, BQ);
      s_kernel<<<gs, 256, 0, stream>>>(embedB, parentB, Sbuf, N);
    }
    {
      dim3 gt(GQ, 1, BQ);
      treefilter_kernel<<<gt, 32, 0, stream>>>(last, Sbuf, orderB, parentB, aggr, filt, N);
    }
    // fusion = latent + filt (in place)
    {
      int total = BQ * CBQ * N;
      add_kernel<<<(total + 255) / 256, 256, 0, stream>>>(lat, filt, total);
    }
    // refined = refine_residual(fusion)
    gemm1x1_wmma<<<g1, blk, 0, stream>>>(Rw1[idx], lat, xbuf, CBQ, CBQ, N, 0,
                                         nullptr, nullptr, nullptr);
    gemm3x3_wmma<<<g1, blk, 0, stream>>>(Rcw[idx], xbuf, t1, CBQ, CBQ, H, Wd, 1,
                                         Rcg[idx], Rcb[idx], nullptr, zpage);
    gemm3x3_wmma<<<g1, blk, 0, stream>>>(Rw2[idx], t1, ref, CBQ, CBQ, H, Wd, 2,
                                         nullptr, nullptr, xbuf, zpage);
    // last_fm = upsample_ac(refined, 2H, 2W)
    {
      int Ho = 2 * H, Wo = 2 * Wd;
      int total = BQ * CBQ * Ho * Wo;
      upsample_kernel<<<(total + 255) / 256, 256, 0, stream>>>(ref, last, H, Wd, Ho, Wo);
    }
  }

  // ---- head2 (only head that reaches the output) on last_fm (96x320) ----
  int Hh = 96, Wh = 320, Nh = Hh * Wh;
  dim3 gh((Nh + 15) / 16, CBQ / (16 * MBLK), BQ);
  gemm1x1_wmma<<<gh, 32, 0, stream>>>(h2_w1, last, xbuf, CBQ, CBQ, Nh, 0,
                                      nullptr, nullptr, nullptr);
  gemm3x3_wmma<<<gh, 32, 0, stream>>>(h2_cbr_w, xbuf, t1, CBQ, CBQ, Hh, Wh, 1,
                                      h2_cbr_g, h2_cbr_b, nullptr, zpage);
  gemm3x3_wmma<<<gh, 32, 0, stream>>>(h2_w2, t1, hbuf, CBQ, CBQ, Hh, Wh, 2,
                                      nullptr, nullptr, xbuf, zpage);
  {
    int total = BQ * Nh;
    head_kernel<<<(total + 255) / 256, 256, 0, stream>>>(hbuf, h2_w, h2_b,
                                                         (float*)d_out, Nh);
  }
}